// RaycastInterpolateFeatures_42597485641918
// MI455X (gfx1250) — compile-verified
//
#include <hip/hip_runtime.h>
#include <hip/hip_bf16.h>

// ---------------- problem constants (from reference) ----------------
#define W_IMG   320
#define H_IMG   240
#define N_STEPS 156
#define DEPTH_MIN_F 2.0f      // 0.1 / 0.05
#define STEP_F      0.5f
#define DXv 192
#define DYv 192
#define DZv 96
#define C_FEAT 128
#define IGNORE_V (-1)

typedef float v2f __attribute__((ext_vector_type(2)));
typedef float v8f __attribute__((ext_vector_type(8)));

// ---------------- kernel 0: zero the segment-sum region ----------------
__global__ void zero_map_kernel(float* __restrict__ map, int n) {
    int i = blockIdx.x * blockDim.x + threadIdx.x;
    if (i < n) map[i] = 0.0f;
}

// ---------------- kernel 1: ray march + weights/indices + histogram ----------------
__device__ __forceinline__ int occ_lookup(const int* __restrict__ occ, int cx, int cy, int cz) {
    int px = min(max(cx, 0), DXv - 1);
    int py = min(max(cy, 0), DYv - 1);
    int pz = min(max(cz, 0), DZv - 1);
    return occ[(pz * DYv + py) * DXv + px];
}

__global__ void raycast_kernel(const float* __restrict__ vm,
                               const float* __restrict__ intr,
                               const int*   __restrict__ occ,
                               float* __restrict__ out_idx,   // n_rays*8 (index as float, -1 masked)
                               float* __restrict__ out_w,     // n_rays*8 normalized weights
                               float* __restrict__ map,       // n_vox, atomic counts
                               int n_rays, int n_vox)
{
    int ray = blockIdx.x * blockDim.x + threadIdx.x;
    if (ray >= n_rays) return;

    int w  = ray % W_IMG;
    int h  = (ray / W_IMG) % H_IMG;
    int vv = ray / (W_IMG * H_IMG);

    const float* M = vm   + vv * 16;   // 4x4 row-major
    const float* K = intr + vv * 4;
    float fx = K[0], fy = K[1], cx = K[2], cy = K[3];
    float dcx = ((float)w + 0.5f - cx) / fx;
    float dcy = ((float)h + 0.5f - cy) / fy;

    // dirs = R * dirs_cam ; origin = t
    float dirx = M[0] * dcx + M[1] * dcy + M[2];
    float diry = M[4] * dcx + M[5] * dcy + M[6];
    float dirz = M[8] * dcx + M[9] * dcy + M[10];
    float ox = M[3], oy = M[7], oz = M[11];

    bool hit = false;
    int  bx = 0, by = 0, bz = 0;
    float frx = 0.f, fry = 0.f, frz = 0.f;

    for (int s = 0; s < N_STEPS; ++s) {
        float d  = DEPTH_MIN_F + STEP_F * (float)s;
        float px = ox + d * dirx - 0.5f;
        float py = oy + d * diry - 0.5f;
        float pz = oz + d * dirz - 0.5f;
        int b0x = (int)floorf(px);
        int b0y = (int)floorf(py);
        int b0z = (int)floorf(pz);
        bool any = false;
        #pragma unroll
        for (int k = 0; k < 8; ++k) {
            int dz = k >> 2, dy = (k >> 1) & 1, dx = k & 1;
            int cxp = b0x + dx, cyp = b0y + dy, czp = b0z + dz;
            bool inb = (cxp >= 0) & (cxp < DXv) & (cyp >= 0) & (cyp < DYv) &
                       (czp >= 0) & (czp < DZv);
            int idx = occ_lookup(occ, cxp, cyp, czp);
            any = any | (inb & (idx != IGNORE_V));
        }
        if (any) {
            hit = true;
            bx = b0x; by = b0y; bz = b0z;
            frx = px - (float)b0x; fry = py - (float)b0y; frz = pz - (float)b0z;
            break;                       // reference keeps first hit only
        }
    }

    // final 8-corner lookup at the recorded base
    float ws[8];
    int   idxs[8];
    bool  valid[8];
    float wsum = 0.f;
    #pragma unroll
    for (int k = 0; k < 8; ++k) {
        int dz = k >> 2, dy = (k >> 1) & 1, dx = k & 1;
        int cxp = bx + dx, cyp = by + dy, czp = bz + dz;
        bool inb = (cxp >= 0) & (cxp < DXv) & (cyp >= 0) & (cyp < DYv) &
                   (czp >= 0) & (czp < DZv);
        int idx = occ_lookup(occ, cxp, cyp, czp);
        idxs[k]  = idx;
        valid[k] = inb & (idx != IGNORE_V);
        float tw = (dz ? frz : 1.f - frz) * (dy ? fry : 1.f - fry) * (dx ? frx : 1.f - frx);
        float wk = (hit && valid[k]) ? tw : 0.f;
        ws[k] = wk;
        wsum += wk;
    }
    float inv = 1.0f / (wsum + 1e-8f);

    long long base8 = (long long)ray * 8;
    #pragma unroll
    for (int k = 0; k < 8; ++k) {
        float nw = ws[k] * inv;
        out_w[base8 + k] = nw;
        int mi = (hit && valid[k]) ? idxs[k] : IGNORE_V;
        out_idx[base8 + k] = (float)mi;
        if (mi != IGNORE_V) {
            int ci = min(max(mi, 0), n_vox - 1);
            atomicAdd(&map[ci], 1.0f);   // exact: sums of 1.0f, deterministic value
        }
    }
}

// ---------------- kernel 2: fp32 WMMA feature projection ----------------
// Per block: 16 rays. P(16x128) = A(16x128) x G(128x128), K split into two
// 64-row passes. A is block-one-hot (ray m owns columns 8m..8m+7 holding its
// normalized trilinear weights), G holds the gathered feature rows in LDS.
// Uses V_WMMA_F32_16X16X4_F32 (full fp32; CDNA5 ISA layouts).
#define GPAD 132
__global__ __launch_bounds__(128)
void project_wmma_kernel(const float* __restrict__ feat,
                         const float* __restrict__ out_idx,
                         const float* __restrict__ out_w,
                         float* __restrict__ proj,
                         int n_vox)
{
    __shared__ float Gs[64][GPAD];   // 64 gathered rows x 128 cols (padded)
    __shared__ float As[16][68];     // weight matrix slice (16 x 64, padded)
    __shared__ int   Ridx[64];       // row -> clamped feature index

    int tid  = threadIdx.x;          // 0..127
    int lane = tid & 31;
    int wave = tid >> 5;             // 0..3
    int ray0 = blockIdx.x * 16;

    int m  = lane & 15;              // A-row / output N within tile reuse
    int n  = lane & 15;              // B/D column within 16-wide N tile
    int kh = lane >> 4;              // lane half selects K {0,1} vs {2,3}, M vs M+8

    int nA = wave * 16;              // first N tile for this wave
    int nB = 64 + wave * 16;         // second N tile

    v8f acc0 = {0.f, 0.f, 0.f, 0.f, 0.f, 0.f, 0.f, 0.f};
    v8f acc1 = {0.f, 0.f, 0.f, 0.f, 0.f, 0.f, 0.f, 0.f};

    for (int pass = 0; pass < 2; ++pass) {
        __syncthreads();             // prior pass's LDS reads are done

        // A slice: A[m][c] nonzero only if ray (pass*8 + c/8) == m
        for (int e = tid; e < 16 * 64; e += 128) {
            int am = e >> 6;
            int ac = e & 63;
            int owner = pass * 8 + (ac >> 3);
            float val = 0.f;
            if (owner == am)
                val = out_w[(long long)(ray0 + am) * 8 + (ac & 7)];
            As[am][ac] = val;
        }
        // row -> feature index (invalid corners have weight 0, clamp to row 0)
        if (tid < 64) {
            int ridx = pass * 64 + tid;
            float fi = out_idx[(long long)(ray0 + (ridx >> 3)) * 8 + (ridx & 7)];
            int ii = (int)fi;
            ii = min(max(ii, 0), n_vox - 1);
            Ridx[tid] = ii;
        }
        __syncthreads();

        // coalesced gather: 128 threads copy one row (512 B) per iteration
        for (int r = 0; r < 64; ++r) {
            Gs[r][tid] = feat[(long long)Ridx[r] * C_FEAT + tid];
        }
        __syncthreads();

        // 16 K-steps of V_WMMA_F32_16X16X4_F32 per N tile
        #pragma unroll
        for (int kk = 0; kk < 16; ++kk) {
            int r0 = 4 * kk + 2 * kh;  // lanes 0-15 -> K {r0=4kk,4kk+1}; lanes 16-31 -> {4kk+2,4kk+3}
            v2f a;  a.x = As[m][r0];          a.y = As[m][r0 + 1];
            v2f b0; b0.x = Gs[r0][nA + n];    b0.y = Gs[r0 + 1][nA + n];
            acc0 = __builtin_amdgcn_wmma_f32_16x16x4_f32(
                       false, a, false, b0, (short)0, acc0, false, false);
            v2f b1; b1.x = Gs[r0][nB + n];    b1.y = Gs[r0 + 1][nB + n];
            acc1 = __builtin_amdgcn_wmma_f32_16x16x4_f32(
                       false, a, false, b1, (short)0, acc1, false, false);
        }
    }

    // D layout: VGPR i holds M = i (lanes 0-15) / M = i+8 (lanes 16-31), N = lane&15
    #pragma unroll
    for (int i = 0; i < 8; ++i) {
        int mrow = i + 8 * kh;
        long long orow = (long long)(ray0 + mrow) * C_FEAT;
        proj[orow + nA + n] = acc0[i];
        proj[orow + nB + n] = acc1[i];
    }
}

// ---------------- launcher ----------------
extern "C" void kernel_launch(void* const* d_in, const int* in_sizes, int n_in,
                              void* d_out, int out_size, void* d_ws, size_t ws_size,
                              hipStream_t stream) {
    const float* feat = (const float*)d_in[0];   // (N_VOX, 128) f32
    const float* vm   = (const float*)d_in[1];   // (B, V, 4, 4) f32
    const float* intr = (const float*)d_in[2];   // (B, V, 4)    f32
    const int*   occ  = (const int*)  d_in[3];   // (B, 96,192,192) i32

    int n_vox   = in_sizes[0] / C_FEAT;
    int n_views = in_sizes[1] / 16;              // B*V
    int n_rays  = n_views * H_IMG * W_IMG;

    float* out = (float*)d_out;
    long long p_proj = 0;
    long long p_idx  = (long long)n_rays * C_FEAT;
    long long p_w    = p_idx + (long long)n_rays * 8;
    long long p_map  = p_w   + (long long)n_rays * 8;

    zero_map_kernel<<<(n_vox + 255) / 256, 256, 0, stream>>>(out + p_map, n_vox);

    raycast_kernel<<<(n_rays + 255) / 256, 256, 0, stream>>>(
        vm, intr, occ, out + p_idx, out + p_w, out + p_map, n_rays, n_vox);

    project_wmma_kernel<<<n_rays / 16, 128, 0, stream>>>(
        feat, out + p_idx, out + p_w, out + p_proj, n_vox);
}